// GATLayer_19155554140769
// MI455X (gfx1250) — compile-verified
//
#include <hip/hip_runtime.h>
#include <math.h>

// ---------------------------------------------------------------------------
// GAT layer for MI455X (gfx1250, wave32).
//  K1: h = X @ W            -> v_wmma_f32_16x16x4_f32, one wave per 16x16 tile
//  K2: s1 = h@a[:d], s2 = h@a[d:]   (wave-per-row dot products)
//  K3/K4: deterministic column-sum of h (for empty softmax rows)
//  K5..K7: CSR-by-target-row build (count / scan / scatter)
//  K8: per-row: sort-by-edge-index (determinism), last-write-wins dedup,
//      softmax over surviving edges, weighted gather of h, ELU.
// ---------------------------------------------------------------------------

typedef __attribute__((ext_vector_type(2))) float v2f;
typedef __attribute__((ext_vector_type(8))) float v8f;

#define LRELU_ALPHA 0.2f
#define ROWCAP 2048

// ---------------- K1: f32 WMMA GEMM  h[N,M] = X[N,K] * W[K,M] ---------------
// Requires N%16==0, M%16==0, K%4==0 (holds: 50000/256/128).
__global__ void gemm_f32_wmma(const float* __restrict__ X,
                              const float* __restrict__ Wm,
                              float* __restrict__ H,
                              int N, int K, int M) {
  const int lane = threadIdx.x & 31;
  const int waveInBlk = threadIdx.x >> 5;
  const int wavesPerBlk = blockDim.x >> 5;
  const int tilesN = M >> 4;                       // tiles along M
  const int tiles = (N >> 4) * tilesN;
  const int tileId = blockIdx.x * wavesPerBlk + waveInBlk;
  if (tileId >= tiles) return;                     // uniform per wave -> EXEC all-1 at WMMA

  const int tileM = tileId / tilesN;
  const int tileN = tileId % tilesN;

  // A 16x4 layout: lanes 0-15 hold M=lane, {K=k,k+1}; lanes 16-31 hold {K=k+2,k+3}
  const int m = (lane & 15);
  const int khalf = (lane >> 4) << 1;              // 0 or 2
  const int row = (tileM << 4) + m;
  const int col = (tileN << 4) + m;                // B/N index for this lane

  const float* __restrict__ xrow = X + (size_t)row * K + khalf;

  v8f acc = {};
  for (int k = 0; k < K; k += 4) {
    v2f a, b;
    a.x = xrow[k + 0];
    a.y = xrow[k + 1];
    // B 4x16 layout (mirror of A): VGPR0 = row (k+khalf), VGPR1 = row (k+khalf+1)
    b.x = Wm[(size_t)(k + khalf)     * M + col];
    b.y = Wm[(size_t)(k + khalf + 1) * M + col];
    acc = __builtin_amdgcn_wmma_f32_16x16x4_f32(false, a, false, b,
                                                (short)0, acc, false, false);
  }
  // C/D layout: VGPR v : lanes 0-15 -> M=v ; lanes 16-31 -> M=8+v ; N=lane&15
  const int r0 = (tileM << 4) + ((lane >> 4) << 3);
#pragma unroll
  for (int v = 0; v < 8; ++v)
    H[(size_t)(r0 + v) * M + col] = acc[v];
}

// ---------------- K2: per-row scores s1 = h.a1, s2 = h.a2 -------------------
__global__ void score_kernel(const float* __restrict__ H,
                             const float* __restrict__ a,
                             float* __restrict__ s1, float* __restrict__ s2,
                             int N, int M) {
  const int wave = (int)((blockIdx.x * blockDim.x + threadIdx.x) >> 5);
  const int lane = threadIdx.x & 31;
  if (wave >= N) return;
  const float* hr = H + (size_t)wave * M;
  float acc1 = 0.f, acc2 = 0.f;
  for (int d = lane; d < M; d += 32) {
    float hv = hr[d];
    acc1 += hv * a[d];
    acc2 += hv * a[M + d];
  }
#pragma unroll
  for (int off = 16; off > 0; off >>= 1) {
    acc1 += __shfl_down(acc1, off, 32);
    acc2 += __shfl_down(acc2, off, 32);
  }
  if (lane == 0) { s1[wave] = acc1; s2[wave] = acc2; }
}

// ---------------- K3/K4: deterministic column sum of h ----------------------
__global__ void colsum_partial(const float* __restrict__ H,
                               float* __restrict__ partial, int N, int M) {
  const int d = threadIdx.x;                 // blockDim.x == M
  float acc = 0.f;
  for (int r = blockIdx.x; r < N; r += gridDim.x)
    acc += H[(size_t)r * M + d];
  partial[(size_t)blockIdx.x * M + d] = acc;
}

__global__ void colsum_reduce(const float* __restrict__ partial,
                              float* __restrict__ colsum, int nParts, int M) {
  const int d = threadIdx.x;
  float acc = 0.f;
  for (int b = 0; b < nParts; ++b)           // fixed order -> deterministic
    acc += partial[(size_t)b * M + d];
  colsum[d] = acc;
}

// ---------------- K5..K7: CSR build -----------------------------------------
__global__ void init_cf(int* __restrict__ counts, int* __restrict__ fill, int T) {
  int i = blockIdx.x * blockDim.x + threadIdx.x;
  if (i < T) { counts[i] = 0; fill[i] = 0; }
}

__global__ void count_kernel(const int* __restrict__ tgt, int* __restrict__ counts,
                             int E, int T) {
  int i = blockIdx.x * blockDim.x + threadIdx.x;
  if (i >= E) return;
  int t = tgt[i];
  if (t >= 0 && t < T) atomicAdd(&counts[t], 1);   // mode="drop"
}

__global__ void scan_kernel(const int* __restrict__ counts,
                            int* __restrict__ rowStart, int T) {
  __shared__ int partial[1024];
  const int tid = threadIdx.x, nth = blockDim.x;
  const int chunk = (T + nth - 1) / nth;
  const int begin = tid * chunk;
  const int end = (begin + chunk < T) ? (begin + chunk) : T;
  int sum = 0;
  for (int i = begin; i < end; ++i) sum += counts[i];
  partial[tid] = sum;
  __syncthreads();
  for (int off = 1; off < nth; off <<= 1) {
    int v = (tid >= off) ? partial[tid - off] : 0;
    __syncthreads();
    partial[tid] += v;
    __syncthreads();
  }
  int excl = (tid == 0) ? 0 : partial[tid - 1];
  for (int i = begin; i < end; ++i) { rowStart[i] = excl; excl += counts[i]; }
  if (tid == 0) rowStart[T] = partial[nth - 1];
}

__global__ void fill_kernel(const int* __restrict__ src, const int* __restrict__ dst,
                            const int* __restrict__ tgt,
                            const float* __restrict__ s1, const float* __restrict__ s2,
                            const int* __restrict__ rowStart, int* __restrict__ fill,
                            int* __restrict__ csrCol, float* __restrict__ csrScore,
                            int* __restrict__ csrIdx, int E, int T) {
  int i = blockIdx.x * blockDim.x + threadIdx.x;
  if (i >= E) return;
  int t = tgt[i];
  if (t < 0 || t >= T) return;
  int sN = src[i], dN = dst[i];
  float e = s1[sN] + s2[dN];
  e = (e > 0.f) ? e : LRELU_ALPHA * e;             // leaky_relu
  int pos = rowStart[t] + atomicAdd(&fill[t], 1);
  csrCol[pos] = dN;
  csrScore[pos] = e;
  csrIdx[pos] = i;
}

// ---------------- K8: per-row softmax + weighted gather ---------------------
__device__ __forceinline__ float block_reduce_max(float v, float* red) {
  const int tid = threadIdx.x;
  red[tid] = v; __syncthreads();
  for (int s = blockDim.x >> 1; s > 0; s >>= 1) {
    if (tid < s) red[tid] = fmaxf(red[tid], red[tid + s]);
    __syncthreads();
  }
  float r = red[0]; __syncthreads();
  return r;
}

__device__ __forceinline__ float block_reduce_sum(float v, float* red) {
  const int tid = threadIdx.x;
  red[tid] = v; __syncthreads();
  for (int s = blockDim.x >> 1; s > 0; s >>= 1) {
    if (tid < s) red[tid] += red[tid + s];
    __syncthreads();
  }
  float r = red[0]; __syncthreads();
  return r;
}

__global__ void row_kernel(const int* __restrict__ rowStart,
                           int* __restrict__ csrCol, float* __restrict__ csrScore,
                           int* __restrict__ csrIdx,
                           const float* __restrict__ H,
                           const float* __restrict__ colsum,
                           float* __restrict__ out,
                           int T, int N, int M) {
  __shared__ int   sc[ROWCAP];
  __shared__ float ss[ROWCAP];
  __shared__ int   si[ROWCAP];
  __shared__ float red[256];

  const int t = blockIdx.x;
  const int tid = threadIdx.x;                      // blockDim.x == M
  const int beg = rowStart[t];
  const int k = rowStart[t + 1] - beg;

  if (k == 0) {
    // full-NEG_FILL row: softmax is uniform 1/N -> mean of h columns
    float v = colsum[tid] / (float)N;
    out[(size_t)t * M + tid] = (v > 0.f) ? v : (expf(v) - 1.f);  // elu
    return;
  }

  float m, sumw;
  if (k <= ROWCAP) {
    for (int j = tid; j < k; j += blockDim.x) {
      sc[j] = csrCol[beg + j]; ss[j] = csrScore[beg + j]; si[j] = csrIdx[beg + j];
    }
    __syncthreads();
    // odd-even transposition sort by original edge index (deterministic order)
    for (int phase = 0; phase < k; ++phase) {
      for (int p = (phase & 1) + 2 * tid; p + 1 < k; p += 2 * blockDim.x) {
        if (si[p] > si[p + 1]) {
          int   ti = si[p]; si[p] = si[p + 1]; si[p + 1] = ti;
          int   tc = sc[p]; sc[p] = sc[p + 1]; sc[p + 1] = tc;
          float tv = ss[p]; ss[p] = ss[p + 1]; ss[p + 1] = tv;
        }
      }
      __syncthreads();
    }
    // last-write-wins dedup: kill j if a later edge hits the same column
    for (int j = tid; j < k; j += blockDim.x) {
      int cj = sc[j];
      for (int l = j + 1; l < k; ++l)
        if (sc[l] == cj) { ss[j] = -INFINITY; break; }
    }
    __syncthreads();
    float lm = -INFINITY;
    for (int j = tid; j < k; j += blockDim.x) lm = fmaxf(lm, ss[j]);
    m = block_reduce_max(lm, red);
    float ls = 0.f;
    for (int j = tid; j < k; j += blockDim.x) {
      float w = expf(ss[j] - m);                     // dead -> exp(-inf)=0
      ss[j] = w; ls += w;
    }
    __syncthreads();
    sumw = block_reduce_sum(ls, red);
    const float inv = 1.f / sumw;
    float acc = 0.f;
    for (int j = 0; j < k; ++j)
      acc += ss[j] * H[(size_t)sc[j] * M + tid];     // h in L2; coalesced over tid
    acc *= inv;
    out[(size_t)t * M + tid] = (acc > 0.f) ? acc : (expf(acc) - 1.f);
  } else {
    // (practically unreachable) global-memory fallback, same math
    for (int j = tid; j < k; j += blockDim.x) {
      int cj = csrCol[beg + j], ij = csrIdx[beg + j];
      bool dead = false;
      for (int l = 0; l < k; ++l)
        if (csrCol[beg + l] == cj && csrIdx[beg + l] > ij) { dead = true; break; }
      if (dead) csrScore[beg + j] = -INFINITY;
    }
    __threadfence_block(); __syncthreads();
    float lm = -INFINITY;
    for (int j = tid; j < k; j += blockDim.x) lm = fmaxf(lm, csrScore[beg + j]);
    m = block_reduce_max(lm, red);
    float ls = 0.f;
    for (int j = tid; j < k; j += blockDim.x) {
      float w = expf(csrScore[beg + j] - m);
      csrScore[beg + j] = w; ls += w;
    }
    __threadfence_block(); __syncthreads();
    sumw = block_reduce_sum(ls, red);
    const float inv = 1.f / sumw;
    float acc = 0.f;
    for (int j = 0; j < k; ++j)
      acc += csrScore[beg + j] * H[(size_t)csrCol[beg + j] * M + tid];
    acc *= inv;
    out[(size_t)t * M + tid] = (acc > 0.f) ? acc : (expf(acc) - 1.f);
  }
}

// ---------------------------------------------------------------------------
extern "C" void kernel_launch(void* const* d_in, const int* in_sizes, int n_in,
                              void* d_out, int out_size, void* d_ws, size_t ws_size,
                              hipStream_t stream) {
  const float* features = (const float*)d_in[0];
  const int*   adj      = (const int*)  d_in[1];   // [2,E]
  const int*   tgt      = (const int*)  d_in[2];   // [E]
  const float* Wm       = (const float*)d_in[3];   // [K,M]
  const float* a        = (const float*)d_in[4];   // [2M,1]

  const int M = in_sizes[4] / 2;                   // 128
  const int K = in_sizes[3] / M;                   // 256
  const int N = in_sizes[0] / K;                   // 50000
  const int E = in_sizes[2];                       // 262144
  const int T = out_size / M;                      // 4096

  const int* src = adj;
  const int* dst = adj + E;

  // workspace carve-out
  char* ws = (char*)d_ws;
  size_t off = 0;
  auto alloc = [&](size_t bytes) -> void* {
    void* p = (void*)(ws + off);
    off += (bytes + 255) & ~(size_t)255;
    return p;
  };
  const int NPARTS = 512;
  float* h        = (float*)alloc((size_t)N * M * sizeof(float));
  float* s1       = (float*)alloc((size_t)N * sizeof(float));
  float* s2       = (float*)alloc((size_t)N * sizeof(float));
  float* partial  = (float*)alloc((size_t)NPARTS * M * sizeof(float));
  float* colsum   = (float*)alloc((size_t)M * sizeof(float));
  int*   counts   = (int*)  alloc((size_t)T * sizeof(int));
  int*   fillc    = (int*)  alloc((size_t)T * sizeof(int));
  int*   rowStart = (int*)  alloc((size_t)(T + 1) * sizeof(int));
  int*   csrCol   = (int*)  alloc((size_t)E * sizeof(int));
  int*   csrIdx   = (int*)  alloc((size_t)E * sizeof(int));
  float* csrScore = (float*)alloc((size_t)E * sizeof(float));
  (void)ws_size; (void)n_in;

  float* out = (float*)d_out;

  // K1: GEMM (one wave32 per 16x16 tile, 4 waves per block)
  {
    const int tiles = (N / 16) * (M / 16);
    const int wavesPerBlk = 4;
    dim3 grid((tiles + wavesPerBlk - 1) / wavesPerBlk), block(32 * wavesPerBlk);
    gemm_f32_wmma<<<grid, block, 0, stream>>>(features, Wm, h, N, K, M);
  }
  // K2: scores (8 waves per block, one wave per row)
  {
    const int wavesPerBlk = 8;
    dim3 grid((N + wavesPerBlk - 1) / wavesPerBlk), block(32 * wavesPerBlk);
    score_kernel<<<grid, block, 0, stream>>>(h, a, s1, s2, N, M);
  }
  // K3/K4: column sums of h (deterministic two-pass)
  colsum_partial<<<dim3(NPARTS), dim3(M), 0, stream>>>(h, partial, N, M);
  colsum_reduce <<<dim3(1),      dim3(M), 0, stream>>>(partial, colsum, NPARTS, M);
  // K5..K7: CSR build
  init_cf     <<<dim3((T + 255) / 256), dim3(256), 0, stream>>>(counts, fillc, T);
  count_kernel<<<dim3((E + 255) / 256), dim3(256), 0, stream>>>(tgt, counts, E, T);
  scan_kernel <<<dim3(1), dim3(1024), 0, stream>>>(counts, rowStart, T);
  fill_kernel <<<dim3((E + 255) / 256), dim3(256), 0, stream>>>(
      src, dst, tgt, s1, s2, rowStart, fillc, csrCol, csrScore, csrIdx, E, T);
  // K8: per-row softmax + gather + ELU
  row_kernel<<<dim3(T), dim3(M), 0, stream>>>(rowStart, csrCol, csrScore, csrIdx,
                                              h, colsum, out, T, N, M);
}